// TSPAutoRegressive_28836410425854
// MI455X (gfx1250) — compile-verified
//
#include <hip/hip_runtime.h>
#include <math.h>

typedef float v2f __attribute__((ext_vector_type(2)));
typedef float v8f __attribute__((ext_vector_type(8)));

#define NEG_INF_F (-1e30f)
#define BN_SCALE  0.99999500003749971893f   /* 1/sqrt(1+1e-5) */
#define INV_8PI   0.039788735772973836f     /* 1/(8*pi) */

__device__ __forceinline__ float wave_max(float v) {
    for (int o = 16; o > 0; o >>= 1) v = fmaxf(v, __shfl_xor(v, o, 32));
    return v;
}
__device__ __forceinline__ float wave_sum(float v) {
    for (int o = 16; o > 0; o >>= 1) v += __shfl_xor(v, o, 32);
    return v;
}

// ---------------------------------------------------------------------------
// Kernel B: fold q = (pos@w_emb.T+b_emb)@Wq.T+bq into Wk per head.
// Wpad (B,16,128): rows 0..7 = 0.25 * sum_j qh[16h+j]*Wk[16h+j,:], rows 8..15 = 0
// cvec (B,8):      0.25 * sum_j qh[16h+j]*bk[16h+j]
// ---------------------------------------------------------------------------
__global__ void prep_kernel(const float* __restrict__ position,
                            const float* __restrict__ w_emb,
                            const float* __restrict__ b_emb,
                            const float* __restrict__ in_proj_w,
                            const float* __restrict__ in_proj_b,
                            float* __restrict__ Wpad,
                            float* __restrict__ cvec) {
    const int b = blockIdx.x;
    const int t = threadIdx.x;  // 128 threads
    __shared__ float qs[128];
    __shared__ float qh[128];
    const float px = position[b * 2 + 0], py = position[b * 2 + 1];
    qs[t] = w_emb[t * 2 + 0] * px + w_emb[t * 2 + 1] * py + b_emb[t];
    __syncthreads();
    float acc = in_proj_b[t];
    for (int j = 0; j < 128; ++j) acc += in_proj_w[t * 128 + j] * qs[j];
    qh[t] = acc;
    __syncthreads();
    if (t < 8) {
        float c = 0.f;
        for (int j = 0; j < 16; ++j)
            c += qh[t * 16 + j] * in_proj_b[128 + t * 16 + j];
        cvec[b * 8 + t] = 0.25f * c;
    }
    for (int h = 0; h < 8; ++h) {
        float s = 0.f;
        for (int j = 0; j < 16; ++j)
            s += qh[h * 16 + j] * in_proj_w[(size_t)(128 + h * 16 + j) * 128 + t];
        Wpad[((size_t)b * 16 + h) * 128 + t] = 0.25f * s;
    }
    for (int h = 8; h < 16; ++h) Wpad[((size_t)b * 16 + h) * 128 + t] = 0.f;
}

// ---------------------------------------------------------------------------
// Kernel C: scores[b,n,h] = emb[b,n,:] . Wpad[b,h,:] + cvec[b,h]
// One wave per 16-row tile; V_WMMA_F32_16X16X4_F32, K-loop of 32.
// HBM-bound: one streaming pass over 268 MB of embeddings.
// ---------------------------------------------------------------------------
__global__ void scores_kernel(const float* __restrict__ emb,
                              const float* __restrict__ Wpad,
                              const float* __restrict__ cvec,
                              float* __restrict__ scores, int N) {
    const int wid = blockIdx.x * 8 + (threadIdx.x >> 5);
    const int lane = threadIdx.x & 31;
    const int tilesPerB = N >> 4;
    const int b = wid / tilesPerB;
    const int n0 = (wid % tilesPerB) << 4;
    const int m = lane & 15;
    const int sel = lane >> 4;

    const float* ap = emb + ((size_t)b * N + n0 + m) * 128 + 2 * sel;   // A: 16x4 rows of emb
    const float* bp = Wpad + ((size_t)b * 16 + m) * 128 + 2 * sel;      // B: 4x16, col = head

    v8f acc = {0.f, 0.f, 0.f, 0.f, 0.f, 0.f, 0.f, 0.f};
#pragma unroll
    for (int k = 0; k < 128; k += 4) {
        v2f a = *(const v2f*)(ap + k);
        v2f bm = *(const v2f*)(bp + k);
        acc = __builtin_amdgcn_wmma_f32_16x16x4_f32(
            false, a, false, bm, (short)0, acc, false, false);
    }
    const int head = lane & 15;
    if (head < 8) {
        const float c = cvec[b * 8 + head];
#pragma unroll
        for (int r = 0; r < 8; ++r) {
            const int n = n0 + r + 8 * sel;  // D: VGPR r -> M=r (lanes<16), M=8+r (lanes>=16)
            scores[((size_t)b * N + n) * 8 + head] = acc[r] + c;
        }
    }
}

// ---------------------------------------------------------------------------
// Kernel D: per-head softmax over N, mean over heads, log-softmax + argmax,
// gather goal coords. One 256-thread block per batch; scores staged in LDS.
// ---------------------------------------------------------------------------
__global__ void node_select_kernel(const float* __restrict__ scores,
                                   const unsigned char* __restrict__ mask_nodes,
                                   const float* __restrict__ node_coords,
                                   int* __restrict__ next_node,
                                   float* __restrict__ goal,
                                   float* __restrict__ lp_node,
                                   float* __restrict__ d_out, int N) {
    const int b = blockIdx.x;
    const int tid = threadIdx.x;  // 256 threads = 8 waves
    __shared__ float sc[1024 * 8];
    __shared__ float npol[1024];
    __shared__ float mh[8], sh[8];
    __shared__ float redv[256];
    __shared__ int redi[256];

    for (int t = tid; t < N * 8; t += 256) {
        const int n = t >> 3;
        float v = scores[(size_t)b * N * 8 + t];
        if (mask_nodes[(size_t)b * N + n]) v = NEG_INF_F;
        sc[t] = v;
    }
    __syncthreads();

    // per-head max + sum(exp) : one wave per head
    const int h = tid >> 5, lane = tid & 31;
    float pm = -3.0e38f;
    for (int n = lane; n < N; n += 32) pm = fmaxf(pm, sc[n * 8 + h]);
    pm = wave_max(pm);
    float ps = 0.f;
    for (int n = lane; n < N; n += 32) ps += expf(sc[n * 8 + h] - pm);
    ps = wave_sum(ps);
    if (lane == 0) { mh[h] = pm; sh[h] = ps; }
    __syncthreads();

    // node_pol[n] = mean_h softmax
    for (int n = tid; n < N; n += 256) {
        float a = 0.f;
#pragma unroll
        for (int hh = 0; hh < 8; ++hh) a += expf(sc[n * 8 + hh] - mh[hh]) / sh[hh];
        float v = 0.125f * a;
        if (mask_nodes[(size_t)b * N + n]) v = NEG_INF_F;
        npol[n] = v;
    }
    __syncthreads();

    // argmax (first occurrence) + max
    float lm = -3.0e38f;
    int li = 0x7fffffff;
    for (int n = tid; n < N; n += 256) {
        const float v = npol[n];
        if (v > lm || (v == lm && n < li)) { lm = v; li = n; }
    }
    redv[tid] = lm; redi[tid] = li;
    __syncthreads();
    for (int s = 128; s > 0; s >>= 1) {
        if (tid < s) {
            const float v2 = redv[tid + s]; const int i2 = redi[tid + s];
            if (v2 > redv[tid] || (v2 == redv[tid] && i2 < redi[tid])) {
                redv[tid] = v2; redi[tid] = i2;
            }
        }
        __syncthreads();
    }
    const float M2 = redv[0];
    const int idx = redi[0];
    __syncthreads();

    // log-softmax normalizer over node_pol (treated as logits by reference)
    float pe = 0.f;
    for (int n = tid; n < N; n += 256) pe += expf(npol[n] - M2);
    redv[tid] = pe;
    __syncthreads();
    for (int s = 128; s > 0; s >>= 1) {
        if (tid < s) redv[tid] += redv[tid + s];
        __syncthreads();
    }
    if (tid == 0) {
        lp_node[b] = npol[idx] - M2 - logf(redv[0]);
        next_node[b] = idx;
        goal[b * 2 + 0] = node_coords[((size_t)b * N + idx) * 2 + 0];
        goal[b * 2 + 1] = node_coords[((size_t)b * N + idx) * 2 + 1];
        d_out[2 * b] = (float)idx;
    }
}

// ---------------------------------------------------------------------------
// Kernel A: 16x16 obstacle-map patch around position. Only the patch is ever
// read, and normalization max_g is attained at the nearest grid point, so
//   w = exp(-(d - d_nearest)/(2 s^2))   (1/(2 pi s^2) cancels).
// ---------------------------------------------------------------------------
__global__ void patch_kernel(const float* __restrict__ position,
                             const float* __restrict__ obstacles,
                             float* __restrict__ patch, int num_obs) {
    const int b = blockIdx.x;
    const int t = threadIdx.x;  // 256
    __shared__ float sobs[64 * 3];
    if (t < num_obs * 3) sobs[t] = obstacles[(size_t)b * num_obs * 3 + t];
    __syncthreads();
    const int a = t >> 4, c = t & 15;  // a = y row, c = x col
    const float px = position[b * 2 + 0], py = position[b * 2 + 1];
    const int map_x = min(max((int)floorf(px * 64.f), 0), 63);
    const int map_y = min(max((int)floorf(py * 64.f), 0), 63);
    const int ix = map_x + c - 8, iy = map_y + a - 8;
    float val = 0.f;
    if (ix >= 0 && ix < 64 && iy >= 0 && iy < 64) {
        const float step = 64.f / 63.f;
        const float lx = ix * step, ly = iy * step;
        for (int o = 0; o < num_obs; ++o) {
            const float x0 = sobs[o * 3 + 0] * 64.f;
            const float y0 = sobs[o * 3 + 1] * 64.f;
            const float s = sobs[o * 3 + 2] * 64.f + 0.01f;
            const int ixn = min(max((int)floorf(x0 * (63.f / 64.f) + 0.5f), 0), 63);
            const int iyn = min(max((int)floorf(y0 * (63.f / 64.f) + 0.5f), 0), 63);
            const float dxn = ixn * step - x0, dyn = iyn * step - y0;
            const float dn = dxn * dxn + dyn * dyn;
            const float dx = lx - x0, dy = ly - y0;
            const float d = dx * dx + dy * dy;
            val += expf(-(d - dn) / (2.f * s * s));
        }
    }
    patch[(size_t)b * 256 + a * 16 + c] = val;
}

// ---------------------------------------------------------------------------
// Kernel E: build 8-channel local maps, conv1+relu+bn, conv2+relu+bn,
// fc1+relu+bn, fc2, masked log-softmax + argmax over 4 dirs, final outputs.
// One 128-thread block per batch; all activations in LDS.
// ---------------------------------------------------------------------------
__global__ void head_kernel(const float* __restrict__ position,
                            const float* __restrict__ patch,
                            const float* __restrict__ goal,
                            const float* __restrict__ lp_node,
                            const unsigned char* __restrict__ mask_dirs,
                            const float* __restrict__ conv1_w, const float* __restrict__ conv1_b,
                            const float* __restrict__ bn1_g, const float* __restrict__ bn1_b,
                            const float* __restrict__ conv2_w, const float* __restrict__ conv2_b,
                            const float* __restrict__ bn2_g, const float* __restrict__ bn2_b,
                            const float* __restrict__ fc1_w, const float* __restrict__ fc1_b,
                            const float* __restrict__ bn3_g, const float* __restrict__ bn3_b,
                            const float* __restrict__ fc2_w, const float* __restrict__ fc2_b,
                            float* __restrict__ d_out, int Btot) {
    const int b = blockIdx.x;
    const int t = threadIdx.x;  // 128
    __shared__ float P[256];
    __shared__ float maps[1024];  // [8][16][8]
    __shared__ float h1[512];     // [4][16][8]
    __shared__ float h2[512];
    __shared__ float hf[128];
    __shared__ float w1s[288];
    __shared__ float w2s[144];
    __shared__ float dlog[4];

    P[t] = patch[(size_t)b * 256 + t];
    P[t + 128] = patch[(size_t)b * 256 + t + 128];
    for (int i = t; i < 288; i += 128) w1s[i] = conv1_w[i];
    for (int i = t; i < 144; i += 128) w2s[i] = conv2_w[i];

    const float px = position[b * 2 + 0], py = position[b * 2 + 1];
    const int map_x = min(max((int)floorf(px * 64.f), 0), 63);
    const int map_y = min(max((int)floorf(py * 64.f), 0), 63);
    const int gxi = min(max((int)floorf(goal[b * 2 + 0] * 64.f), map_x - 8), map_x + 8) - map_x + 8;
    const int gyi = min(max((int)floorf(goal[b * 2 + 1] * 64.f), map_y - 8), map_y + 8) - map_y + 8;
    __syncthreads();

    // maps: ch 0..3 = obs {east,north,west,south}; ch 4..7 = goal gaussians
    for (int idx = t; idx < 1024; idx += 128) {
        const int ch = idx >> 7, i = (idx >> 3) & 15, j = idx & 7;
        float v;
        if (ch == 0)      v = P[i * 16 + 8 + j];
        else if (ch == 1) v = P[(8 + j) * 16 + i];
        else if (ch == 2) v = P[i * 16 + j];
        else if (ch == 3) v = P[j * 16 + i];
        else {
            int xx, yy;
            if (ch == 4)      { xx = i;     yy = 8 + j; }
            else if (ch == 5) { xx = 8 + j; yy = i;     }
            else if (ch == 6) { xx = i;     yy = j;     }
            else              { xx = j;     yy = i;     }
            const float dx = (float)(xx - gxi), dy = (float)(yy - gyi);
            v = INV_8PI * expf(-(dx * dx + dy * dy) * 0.125f);
        }
        maps[ch * 128 + i * 8 + j] = v;
    }
    __syncthreads();

    // conv1 (8->4, 3x3 SAME on 16x8) + relu + bn
    for (int idx = t; idx < 512; idx += 128) {
        const int o = idx >> 7, r = (idx >> 3) & 15, cc = idx & 7;
        float acc = conv1_b[o];
        for (int ic = 0; ic < 8; ++ic)
            for (int ky = 0; ky < 3; ++ky) {
                const int rr = r + ky - 1;
                if (rr < 0 || rr > 15) continue;
                for (int kx = 0; kx < 3; ++kx) {
                    const int cx = cc + kx - 1;
                    if (cx < 0 || cx > 7) continue;
                    acc += maps[ic * 128 + rr * 8 + cx] * w1s[((o * 8 + ic) * 3 + ky) * 3 + kx];
                }
            }
        acc = fmaxf(acc, 0.f);
        h1[idx] = acc * (bn1_g[o] * BN_SCALE) + bn1_b[o];
    }
    __syncthreads();

    // conv2 (4->4) + relu + bn
    for (int idx = t; idx < 512; idx += 128) {
        const int o = idx >> 7, r = (idx >> 3) & 15, cc = idx & 7;
        float acc = conv2_b[o];
        for (int ic = 0; ic < 4; ++ic)
            for (int ky = 0; ky < 3; ++ky) {
                const int rr = r + ky - 1;
                if (rr < 0 || rr > 15) continue;
                for (int kx = 0; kx < 3; ++kx) {
                    const int cx = cc + kx - 1;
                    if (cx < 0 || cx > 7) continue;
                    acc += h1[ic * 128 + rr * 8 + cx] * w2s[((o * 4 + ic) * 3 + ky) * 3 + kx];
                }
            }
        acc = fmaxf(acc, 0.f);
        h2[idx] = acc * (bn2_g[o] * BN_SCALE) + bn2_b[o];
    }
    __syncthreads();

    // fc1 (512->128) + relu + bn1d
    {
        float acc = fc1_b[t];
        const float* wr = fc1_w + (size_t)t * 512;
        for (int k = 0; k < 512; ++k) acc += h2[k] * wr[k];
        acc = fmaxf(acc, 0.f);
        hf[t] = acc * (bn3_g[t] * BN_SCALE) + bn3_b[t];
    }
    __syncthreads();

    // fc2 (128->4), mask, log-softmax + argmax, final outputs
    if (t < 4) {
        float acc = fc2_b[t];
        const float* wr = fc2_w + (size_t)t * 128;
        for (int k = 0; k < 128; ++k) acc += hf[k] * wr[k];
        if (mask_dirs[b * 4 + t]) acc = NEG_INF_F;
        dlog[t] = acc;
    }
    __syncthreads();
    if (t == 0) {
        float m = dlog[0];
        int dir = 0;
        for (int o = 1; o < 4; ++o)
            if (dlog[o] > m) { m = dlog[o]; dir = o; }
        float se = 0.f;
        for (int o = 0; o < 4; ++o) se += expf(dlog[o] - m);
        const float lp_dir = dlog[dir] - m - logf(se);
        d_out[2 * b + 1] = (float)dir;
        d_out[2 * Btot + b] = 0.5f * (lp_node[b] + lp_dir);
    }
}

// ---------------------------------------------------------------------------
extern "C" void kernel_launch(void* const* d_in, const int* in_sizes, int n_in,
                              void* d_out, int out_size, void* d_ws, size_t ws_size,
                              hipStream_t stream) {
    const float* emb        = (const float*)d_in[0];
    const float* position   = (const float*)d_in[1];
    const float* obstacles  = (const float*)d_in[2];
    const float* node_coords= (const float*)d_in[3];
    const unsigned char* mask_nodes = (const unsigned char*)d_in[4];
    const unsigned char* mask_dirs  = (const unsigned char*)d_in[5];
    const float* w_emb     = (const float*)d_in[6];
    const float* b_emb     = (const float*)d_in[7];
    const float* in_proj_w = (const float*)d_in[8];
    const float* in_proj_b = (const float*)d_in[9];
    const float* conv1_w = (const float*)d_in[10];
    const float* conv1_b = (const float*)d_in[11];
    const float* bn1_g   = (const float*)d_in[12];
    const float* bn1_b   = (const float*)d_in[13];
    const float* conv2_w = (const float*)d_in[14];
    const float* conv2_b = (const float*)d_in[15];
    const float* bn2_g   = (const float*)d_in[16];
    const float* bn2_b   = (const float*)d_in[17];
    const float* fc1_w   = (const float*)d_in[18];
    const float* fc1_b   = (const float*)d_in[19];
    const float* bn3_g   = (const float*)d_in[20];
    const float* bn3_b   = (const float*)d_in[21];
    const float* fc2_w   = (const float*)d_in[22];
    const float* fc2_b   = (const float*)d_in[23];
    float* out = (float*)d_out;

    const int B = in_sizes[1] / 2;                 // 512
    const int H = 128;
    const int N = in_sizes[0] / (B * H);           // 1024
    const int num_obs = in_sizes[2] / (B * 3);     // 20

    float* ws = (float*)d_ws;
    size_t off = 0;
    float* scores = ws + off; off += (size_t)B * N * 8;
    float* Wpad   = ws + off; off += (size_t)B * 16 * 128;
    float* cvec   = ws + off; off += (size_t)B * 8;
    float* patch  = ws + off; off += (size_t)B * 256;
    int*   nnode  = (int*)(ws + off); off += (size_t)B;
    float* goal   = ws + off; off += (size_t)B * 2;
    float* lpn    = ws + off; off += (size_t)B;

    prep_kernel<<<B, 128, 0, stream>>>(position, w_emb, b_emb, in_proj_w, in_proj_b,
                                       Wpad, cvec);
    scores_kernel<<<(B * (N / 16)) / 8, 256, 0, stream>>>(emb, Wpad, cvec, scores, N);
    node_select_kernel<<<B, 256, 0, stream>>>(scores, mask_nodes, node_coords,
                                              nnode, goal, lpn, out, N);
    patch_kernel<<<B, 256, 0, stream>>>(position, obstacles, patch, num_obs);
    head_kernel<<<B, 128, 0, stream>>>(position, patch, goal, lpn, mask_dirs,
                                       conv1_w, conv1_b, bn1_g, bn1_b,
                                       conv2_w, conv2_b, bn2_g, bn2_b,
                                       fc1_w, fc1_b, bn3_g, bn3_b,
                                       fc2_w, fc2_b, out, B);
}